// Critias_54443005444198
// MI455X (gfx1250) — compile-verified
//
#include <hip/hip_runtime.h>
#include <math.h>

typedef float v2f __attribute__((ext_vector_type(2)));
typedef float v8f __attribute__((ext_vector_type(8)));
typedef unsigned int v4u __attribute__((ext_vector_type(4)));
typedef int v4i __attribute__((ext_vector_type(4)));
typedef int v8i __attribute__((ext_vector_type(8)));

#define NPAIR 100000
#define NNODE 10000

// ---- workspace layout (float offsets) ----
#define OFF_SPECIES   0            // NNODE*4
#define OFF_NODESTART 40000        // (NNODE+1) ints, padded to 10016 slots
#define OFF_R         50016        // NPAIR
#define OFF_CUT       150016       // NPAIR
#define OFF_PS        250016       // NPAIR
#define OFF_NBR       350016       // NNODE
#define OFF_SPH       360016       // NPAIR*25  (raw solid harmonics)
#define OFF_RADT      2860016      // 64*NPAIR  (radial, transposed [r][p])
#define OFF_C1        9260016      // NPAIR*16
#define OFF_SC0       10860016     // NNODE*16
#define OFF_C2        11020016     // NPAIR*50
#define OFF_SPHF      16020016     // NNODE*250
#define OFF_SC1       18520016     // NNODE*16
#define OFF_C3        18680016     // NPAIR*16
// total ~20.3M floats ~ 81 MB

__device__ __forceinline__ v8f wmma4(v2f a, v2f b, v8f c) {
  return __builtin_amdgcn_wmma_f32_16x16x4_f32(false, a, false, b, (short)0, c, false, false);
}
__device__ __forceinline__ v8f z8() {
  v8f v = {0.f,0.f,0.f,0.f,0.f,0.f,0.f,0.f};
  return v;
}

// Tensor Data Mover: 2D tile load Global -> LDS (D# per CDNA5 ISA ch.8).
// td0/td1: tensor dims (elements), tile0/tile1: tile dims, stride0: dim0 stride.
// data_size = 4 bytes, no iterate/pad/gather, workgroup_mask = 0 (not in cluster).
// This toolchain carries the 6-arg builtin: (v4u, v8i, v4i, v4i, v8i, i32 cpol).
__device__ __forceinline__ void tdm_load_2d(unsigned lds_addr, const void* gsrc,
                                            unsigned td0, unsigned td1,
                                            unsigned tile0, unsigned tile1,
                                            unsigned stride0) {
  unsigned long long ga = (unsigned long long)(size_t)gsrc;
  v4u g0;
  g0.x = 1u;                                               // count=1 (valid user D#)
  g0.y = lds_addr;                                         // LDS byte address
  g0.z = (unsigned)ga;                                     // global_addr[31:0]
  g0.w = (unsigned)((ga >> 32) & 0x1FFFFFFu) | (2u << 30); // global_addr[56:32] | type=2
  v8i g1;
  g1[0] = (int)(2u << 16);                                 // data_size code 2 = 4B
  g1[1] = (int)((td0 & 0xFFFFu) << 16);                    // tensor_dim0[15:0]
  g1[2] = (int)(((td0 >> 16) & 0xFFFFu) | ((td1 & 0xFFFFu) << 16));
  g1[3] = (int)(((td1 >> 16) & 0xFFFFu) | ((tile0 & 0xFFFFu) << 16));
  g1[4] = (int)(tile1 & 0xFFFFu);                          // tile_dim1, tile_dim2=0
  g1[5] = (int)stride0;                                    // tensor_dim0_stride[31:0]
  g1[6] = 0;
  g1[7] = 0;
  v4i gz4 = {0, 0, 0, 0};
  v8i gz8 = {0, 0, 0, 0, 0, 0, 0, 0};
  __builtin_amdgcn_tensor_load_to_lds(g0, g1, gz4, gz4, gz8, 0);
}

// ---------------- species embedding ----------------
__global__ void k_species(const int* __restrict__ Z, const unsigned char* __restrict__ nm,
                          const float* __restrict__ embed, float* __restrict__ ws) {
  int n = blockIdx.x * blockDim.x + threadIdx.x;
  if (n >= NNODE) return;
  float s = nm[n] ? 1.f : 0.f;
  int z = Z[n];
  for (int c = 0; c < 4; ++c) ws[OFF_SPECIES + n*4 + c] = embed[z*4 + c] * s;
}

// ---------------- node row offsets (i is sorted) ----------------
__global__ void k_nodestart(const int* __restrict__ gi, float* __restrict__ ws) {
  int n = blockIdx.x * blockDim.x + threadIdx.x;
  if (n > NNODE) return;
  int lo = 0, hi = NPAIR;
  while (lo < hi) { int mid = (lo + hi) >> 1; if (gi[mid] < n) lo = mid + 1; else hi = mid; }
  ((int*)(ws + OFF_NODESTART))[n] = lo;
}

// ---------------- per-pair geometry: r, cut, sph, radial (transposed) ----------------
__global__ void k_pair(const float* __restrict__ R, const unsigned char* __restrict__ pmask,
                       float* __restrict__ ws) {
  __shared__ float lb[64];
  if (threadIdx.x < 64) {
    float v = (float)threadIdx.x;
    lb[threadIdx.x] = lgammaf(64.f) - lgammaf(v + 1.f) - lgammaf(64.f - v);
  }
  __syncthreads();
  int p = blockIdx.x * blockDim.x + threadIdx.x;
  if (p >= NPAIR) return;
  float x = R[p*3+0], y = R[p*3+1], z = R[p*3+2];
  float pm = pmask[p] ? 1.f : 0.f;
  float x2 = x*x, y2 = y*y, z2 = z*z;
  float r2 = x2 + y2 + z2;
  float r = sqrtf(r2 + 1e-12f);
  ws[OFF_R + p] = r;
  float cut = 0.f;
  if (r < 5.f) cut = 0.5f * (cosf(3.14159265358979323846f * r * 0.2f) + 1.f);
  cut *= pm;
  ws[OFF_CUT + p] = cut;

  const float S3 = 1.7320508075688772f, S5 = 2.23606797749979f, S10 = 3.1622776601683795f;
  const float S15 = 3.872983346207417f, S35 = 5.916079783099616f, S70 = 8.366600265340756f;
  const float S58 = 0.7905694150420949f, S38 = 0.6123724356957945f;
  float sp[25];
  sp[0] = 1.f; sp[1] = y; sp[2] = z; sp[3] = x;
  sp[4] = S3*x*y; sp[5] = S3*y*z; sp[6] = 0.5f*(3.f*z2 - r2); sp[7] = S3*x*z; sp[8] = 0.5f*S3*(x2 - y2);
  sp[9]  = S58*y*(3.f*x2 - y2); sp[10] = S15*x*y*z; sp[11] = S38*y*(5.f*z2 - r2);
  sp[12] = 0.5f*z*(5.f*z2 - 3.f*r2); sp[13] = S38*x*(5.f*z2 - r2); sp[14] = 0.5f*S15*z*(x2 - y2);
  sp[15] = S58*x*(x2 - 3.f*y2);
  sp[16] = 0.5f*S35*x*y*(x2 - y2); sp[17] = 0.25f*S70*y*z*(3.f*x2 - y2);
  sp[18] = 0.5f*S5*x*y*(7.f*z2 - r2); sp[19] = 0.25f*S10*y*z*(7.f*z2 - 3.f*r2);
  sp[20] = 0.125f*(35.f*z2*z2 - 30.f*z2*r2 + 3.f*r2*r2);
  sp[21] = 0.25f*S10*x*z*(7.f*z2 - 3.f*r2); sp[22] = 0.25f*S5*(x2 - y2)*(7.f*z2 - r2);
  sp[23] = 0.25f*S70*x*z*(x2 - 3.f*y2); sp[24] = 0.125f*S35*(x2*x2 - 6.f*x2*y2 + y2*y2);
  for (int k = 0; k < 25; ++k) ws[OFF_SPH + p*25 + k] = sp[k];

  float xc = r * 0.2f;
  xc = fminf(fmaxf(xc, 1e-7f), 1.f - 1e-7f);
  float lx = logf(xc), l1x = log1pf(-xc);
  float* radT = ws + OFF_RADT;
  for (int v = 0; v < 64; ++v) {
    float fv = (float)v;
    radT[v*NPAIR + p] = expf(lb[v] + fv*lx + (63.f - fv)*l1x) * pm;
  }
}

// ---------------- neighbor count per node ----------------
__global__ void k_nbr(float* __restrict__ ws) {
  int n = blockIdx.x * blockDim.x + threadIdx.x;
  if (n >= NNODE) return;
  const int* ns = (const int*)(ws + OFF_NODESTART);
  float s = 0.f;
  for (int p = ns[n]; p < ns[n+1]; ++p) s += ws[OFF_CUT + p];
  ws[OFF_NBR + n] = s + 1.f;
}

// ---------------- pair scale ----------------
__global__ void k_ps(const int* __restrict__ gi, const unsigned char* __restrict__ pmask,
                     float* __restrict__ ws) {
  int p = blockIdx.x * blockDim.x + threadIdx.x;
  if (p >= NPAIR) return;
  ws[OFF_PS + p] = pmask[p] ? (1.f / ws[OFF_NBR + gi[p]]) : 0.f;
}

// ---------------- stage A: pf(8)+bias -> 64x16 radial-contracted, WMMA, K=768 ----------------
__global__ void __launch_bounds__(256) k_stageA(const int* __restrict__ gi, const int* __restrict__ gj,
                        const float* __restrict__ W1, const float* __restrict__ b1,
                        float* __restrict__ ws) {
  __shared__ float Wl[768 * 16];  // kidx = r*12+k ; k: 0..7=W rows, 8=bias, 9..11=0
  int t = threadIdx.x;
  for (int e = t; e < 768*16; e += 256) {
    int kidx = e >> 4, n = e & 15;
    int r = kidx / 12, k = kidx % 12;
    float v = 0.f;
    if (k < 8)       v = W1[k*1024 + r*16 + n];
    else if (k == 8) v = b1[r*16 + n];
    Wl[e] = v;
  }
  __syncthreads();
  int wave = t >> 5, lane = t & 31;
  int base = (blockIdx.x * 8 + wave) * 16;
  if (base >= NPAIR) return;
  int m = lane & 15, hi = lane >> 4, kb = hi * 2, nl = lane & 15;
  int p = base + m; if (p > NPAIR - 1) p = NPAIR - 1;
  int ii = gi[p], jj = gj[p];
  const float* species = ws + OFF_SPECIES;
  const float* radT = ws + OFF_RADT;
  float pfa[6];
  #pragma unroll
  for (int ks = 0; ks < 3; ++ks) {
    #pragma unroll
    for (int u = 0; u < 2; ++u) {
      int k = 4*ks + kb + u;
      float v;
      if (k < 4)       v = species[ii*4 + k];
      else if (k < 8)  v = species[jj*4 + (k - 4)];
      else if (k == 8) v = 1.f;
      else             v = 0.f;
      pfa[2*ks + u] = v;
    }
  }
  v8f acc = z8();
  for (int r = 0; r < 64; ++r) {
    float rad = radT[r*NPAIR + p];
    if (r < 63) __builtin_prefetch(&radT[(r+1)*NPAIR + p], 0, 1);
    #pragma unroll
    for (int ks = 0; ks < 3; ++ks) {
      v2f a; a.x = pfa[2*ks] * rad; a.y = pfa[2*ks+1] * rad;
      int krow = r*12 + 4*ks + kb;
      v2f b; b.x = Wl[krow*16 + nl]; b.y = Wl[(krow+1)*16 + nl];
      acc = wmma4(a, b, acc);
    }
  }
  const float* cutA = ws + OFF_CUT;
  const float* psA  = ws + OFF_PS;
  float* c1 = ws + OFF_C1;
  #pragma unroll
  for (int v = 0; v < 8; ++v) {
    int row = base + v + 8*hi;
    if (row < NPAIR) {
      float w = cutA[row] * psA[row];
      c1[row*16 + nl] = acc[v] * w;
    }
  }
}

// ---------------- segment sum of c1 -> scalar0 ----------------
__global__ void k_sumA(const unsigned char* __restrict__ nm, float* __restrict__ ws) {
  int t = blockIdx.x * blockDim.x + threadIdx.x;
  if (t >= NNODE * 16) return;
  int n = t >> 4, f = t & 15;
  const int* ns = (const int*)(ws + OFF_NODESTART);
  const float* c1 = ws + OFF_C1;
  float s = 0.f;
  for (int p = ns[n]; p < ns[n+1]; ++p) s += c1[p*16 + f];
  ws[OFF_SC0 + t] = nm[n] ? s : 0.f;
}

// ---------------- stage B: pf2(20) -> 64x50 radial-contracted, WMMA, TDM weight staging ----
// W contribution: K=20 exactly (5 k-steps), raw W2 tile TDM-copied to LDS per 8-r chunk.
// Bias contribution: separate WMMA group with A = radial, B = TDM-copied b2 chunk.
// Columns g>=50 compute garbage but are never stored (GEMM columns are independent).
__global__ void __launch_bounds__(256) k_stageB(const int* __restrict__ gi, const int* __restrict__ gj,
                        const float* __restrict__ W2, const float* __restrict__ b2,
                        float* __restrict__ ws) {
  __shared__ float Wl2[20*400 + 400 + 64];  // [k*400 + col] ; bias at 8000 ; +64 pad for stray cols
  int t = threadIdx.x;
  int wave = t >> 5, lane = t & 31;
  int base = (blockIdx.x * 8 + wave) * 16;
  int m = lane & 15, hi = lane >> 4, kb = hi * 2, nl = lane & 15;
  int p = base + m; if (p > NPAIR - 1) p = NPAIR - 1;
  int ii = gi[p], jj = gj[p];
  const float* species = ws + OFF_SPECIES;
  const float* sc0 = ws + OFF_SC0;
  const float* radT = ws + OFF_RADT;
  float pfb[10];
  #pragma unroll
  for (int ks = 0; ks < 5; ++ks) {
    #pragma unroll
    for (int u = 0; u < 2; ++u) {
      int k = 4*ks + kb + u;
      float v;
      if (k < 16)      v = sc0[ii*16 + k];
      else             v = species[jj*4 + (k - 16)];
      pfb[2*ks + u] = v;
    }
  }
  v8f acc[4];
  #pragma unroll
  for (int nt = 0; nt < 4; ++nt) acc[nt] = z8();

  unsigned ldsW = (unsigned)(size_t)(&Wl2[0]);   // flat LDS addr[31:0] = LDS byte offset
  unsigned ldsB = ldsW + 8000u * 4u;

  for (int c = 0; c < 8; ++c) {
    __syncthreads();                       // previous chunk fully consumed
    if (wave == 0) {
      // rows k=0..19 of W2, columns [c*400, c*400+400) of the 3200-wide tensor
      tdm_load_2d(ldsW, W2 + c*400, 3200u, 20u, 400u, 20u, 3200u);
      // bias chunk: 400 contiguous elements of b2
      tdm_load_2d(ldsB, b2 + c*400, 3200u, 1u, 400u, 1u, 3200u);
      __builtin_amdgcn_s_wait_tensorcnt(0);
    }
    __syncthreads();                       // LDS tile visible to all waves
    for (int rl = 0; rl < 8; ++rl) {
      int rr = c*8 + rl;
      float rad = radT[rr*NPAIR + p];
      int colb = rl*50 + nl;
      #pragma unroll
      for (int ks = 0; ks < 5; ++ks) {
        v2f a; a.x = pfb[2*ks] * rad; a.y = pfb[2*ks+1] * rad;
        int krow = 4*ks + kb;
        #pragma unroll
        for (int nt = 0; nt < 4; ++nt) {
          v2f b; b.x = Wl2[krow*400 + colb + nt*16]; b.y = Wl2[(krow+1)*400 + colb + nt*16];
          acc[nt] = wmma4(a, b, acc[nt]);
        }
      }
    }
    // bias term: out[p,g] += sum_r radial[p,r] * b2[r*50+g]  (A = radial, K = 8 per chunk)
    #pragma unroll
    for (int ks = 0; ks < 2; ++ks) {
      int rowl = 4*ks + kb;
      int r0 = c*8 + rowl;
      v2f a; a.x = radT[r0*NPAIR + p]; a.y = radT[(r0+1)*NPAIR + p];
      #pragma unroll
      for (int nt = 0; nt < 4; ++nt) {
        v2f b; b.x = Wl2[8000 + rowl*50 + nt*16 + nl]; b.y = Wl2[8000 + (rowl+1)*50 + nt*16 + nl];
        acc[nt] = wmma4(a, b, acc[nt]);
      }
    }
  }
  const float* cutA = ws + OFF_CUT;
  float* c2 = ws + OFF_C2;
  #pragma unroll
  for (int nt = 0; nt < 4; ++nt) {
    int g = nt*16 + nl;
    if (g >= 50) continue;
    #pragma unroll
    for (int v = 0; v < 8; ++v) {
      int row = base + v + 8*hi;
      if (row < NPAIR) c2[row*50 + g] = acc[nt][v] * cutA[row];
    }
  }
}

// ---------------- sphf segment sum: sphf[n,lm,s] = sum c2[p,l*10+s]*sph[p,lm]*ps[p] ----------------
__global__ void k_sphf(const unsigned char* __restrict__ nm, float* __restrict__ ws) {
  int t = blockIdx.x * blockDim.x + threadIdx.x;
  if (t >= NNODE * 250) return;
  int n = t / 250, lmS = t % 250;
  int lm = lmS / 10, s = lmS % 10;
  int l = (lm < 1) ? 0 : (lm < 4) ? 1 : (lm < 9) ? 2 : (lm < 16) ? 3 : 4;
  int g = l*10 + s;
  const int* ns = (const int*)(ws + OFF_NODESTART);
  const float* c2 = ws + OFF_C2;
  const float* sph = ws + OFF_SPH;
  const float* psA = ws + OFF_PS;
  float acc = 0.f;
  for (int p = ns[n]; p < ns[n+1]; ++p)
    acc += c2[p*50 + g] * sph[p*25 + lm] * psA[p];
  ws[OFF_SPHF + t] = nm[n] ? acc : 0.f;
}

// ---------------- node head 1: trace, +W_tr, layernorm1, q, scalar1 ----------------
__global__ void k_node1(const unsigned char* __restrict__ nm,
                        const float* __restrict__ Wtr, const float* __restrict__ btr,
                        const float* __restrict__ ln1s, const float* __restrict__ ln1b,
                        const float* __restrict__ Wq1, const float* __restrict__ bq1,
                        const float* __restrict__ Wq2, const float* __restrict__ bq2,
                        float* __restrict__ ws, float* __restrict__ out) {
  int n = blockIdx.x * blockDim.x + threadIdx.x;
  if (n >= NNODE) return;
  const float* sphf = ws + OFF_SPHF + (size_t)n * 250;
  float tr[50];
  int lm0 = 0;
  for (int l = 0; l < 5; ++l) {
    int cnt = 2*l + 1;
    for (int s = 0; s < 10; ++s) {
      float a = 0.f;
      for (int mm = 0; mm < cnt; ++mm) { float v = sphf[(lm0+mm)*10 + s]; a += v*v; }
      tr[l*10 + s] = a;
    }
    lm0 += cnt;
  }
  bool act = nm[n] != 0;
  float sv[16];
  for (int f = 0; f < 16; ++f) sv[f] = ws[OFF_SC0 + n*16 + f];
  if (act) {
    for (int f = 0; f < 16; ++f) {
      float a = btr[f];
      for (int g = 0; g < 50; ++g) a += tr[g] * Wtr[g*16 + f];
      sv[f] += a;
    }
  }
  float o[16];
  float qv = 0.f;
  if (act) {
    float mu = 0.f;
    for (int f = 0; f < 16; ++f) mu += sv[f];
    mu *= (1.f/16.f);
    float var = 0.f;
    for (int f = 0; f < 16; ++f) { float d = sv[f] - mu; var += d*d; }
    var *= (1.f/16.f);
    float inv = rsqrtf(var + 1e-6f);
    for (int f = 0; f < 16; ++f) o[f] = (sv[f] - mu) * inv * ln1s[f] + ln1b[f];
    float accq = bq2[0];
    for (int f2 = 0; f2 < 16; ++f2) {
      float h = bq1[f2];
      for (int f = 0; f < 16; ++f) h += o[f] * Wq1[f*16 + f2];
      h = h / (1.f + expf(-h));  // silu
      accq += h * Wq2[f2];
    }
    qv = accq;
  } else {
    for (int f = 0; f < 16; ++f) o[f] = 0.f;
  }
  for (int f = 0; f < 16; ++f) ws[OFF_SC1 + n*16 + f] = o[f];
  out[NNODE + n] = qv;
}

// ---------------- stage C: pf3(32)+bias -> 64x16, WMMA, K=2304 ----------------
__global__ void __launch_bounds__(256) k_stageC(const int* __restrict__ gi, const int* __restrict__ gj,
                        const float* __restrict__ W3, const float* __restrict__ b3,
                        float* __restrict__ ws) {
  __shared__ float Wl[16 * 36 * 16];  // 16 r per chunk; kidx = rl*36+k
  int t = threadIdx.x;
  int wave = t >> 5, lane = t & 31;
  int base = (blockIdx.x * 8 + wave) * 16;
  int m = lane & 15, hi = lane >> 4, kb = hi * 2, nl = lane & 15;
  int p = base + m; if (p > NPAIR - 1) p = NPAIR - 1;
  int ii = gi[p], jj = gj[p];
  const float* sc1 = ws + OFF_SC1;
  const float* radT = ws + OFF_RADT;
  float pfc[18];
  #pragma unroll
  for (int ks = 0; ks < 9; ++ks) {
    #pragma unroll
    for (int u = 0; u < 2; ++u) {
      int k = 4*ks + kb + u;
      float v;
      if (k < 16)       v = sc1[ii*16 + k];
      else if (k < 32)  v = sc1[jj*16 + (k - 16)];
      else if (k == 32) v = 1.f;
      else              v = 0.f;
      pfc[2*ks + u] = v;
    }
  }
  v8f acc = z8();
  for (int c = 0; c < 4; ++c) {
    __syncthreads();
    for (int e = t; e < 9216; e += 256) {
      int kidx = e >> 4, n = e & 15;
      int rl = kidx / 36, k = kidx % 36;
      int rr = c*16 + rl;
      float v = 0.f;
      if (k < 32)       v = W3[k*1024 + rr*16 + n];
      else if (k == 32) v = b3[rr*16 + n];
      Wl[e] = v;
    }
    __syncthreads();
    for (int rl = 0; rl < 16; ++rl) {
      int rr = c*16 + rl;
      float rad = radT[rr*NPAIR + p];
      #pragma unroll
      for (int ks = 0; ks < 9; ++ks) {
        v2f a; a.x = pfc[2*ks] * rad; a.y = pfc[2*ks+1] * rad;
        int krow = rl*36 + 4*ks + kb;
        v2f b; b.x = Wl[krow*16 + nl]; b.y = Wl[(krow+1)*16 + nl];
        acc = wmma4(a, b, acc);
      }
    }
  }
  const float* cutA = ws + OFF_CUT;
  const float* psA  = ws + OFF_PS;
  float* c3 = ws + OFF_C3;
  #pragma unroll
  for (int v = 0; v < 8; ++v) {
    int row = base + v + 8*hi;
    if (row < NPAIR) {
      float w = cutA[row] * psA[row];
      c3[row*16 + nl] = acc[v] * w;
    }
  }
}

// ---------------- node head 2: d segsum, layernorm2, energy ----------------
__global__ void k_node2(const unsigned char* __restrict__ nm,
                        const float* __restrict__ ln2s, const float* __restrict__ ln2b,
                        const float* __restrict__ We1, const float* __restrict__ be1,
                        const float* __restrict__ We2, const float* __restrict__ be2,
                        float* __restrict__ ws, float* __restrict__ out) {
  int n = blockIdx.x * blockDim.x + threadIdx.x;
  if (n >= NNODE) return;
  const int* ns = (const int*)(ws + OFF_NODESTART);
  const float* c3 = ws + OFF_C3;
  bool act = nm[n] != 0;
  float sv[16];
  for (int f = 0; f < 16; ++f) sv[f] = ws[OFF_SC1 + n*16 + f];
  if (act) {
    float d[16];
    for (int f = 0; f < 16; ++f) d[f] = 0.f;
    for (int p = ns[n]; p < ns[n+1]; ++p)
      for (int f = 0; f < 16; ++f) d[f] += c3[p*16 + f];
    for (int f = 0; f < 16; ++f) sv[f] += d[f];
  }
  float e = 0.f;
  if (act) {
    float mu = 0.f;
    for (int f = 0; f < 16; ++f) mu += sv[f];
    mu *= (1.f/16.f);
    float var = 0.f;
    for (int f = 0; f < 16; ++f) { float dd = sv[f] - mu; var += dd*dd; }
    var *= (1.f/16.f);
    float inv = rsqrtf(var + 1e-6f);
    float o[16];
    for (int f = 0; f < 16; ++f) o[f] = (sv[f] - mu) * inv * ln2s[f] + ln2b[f];
    float acce = be2[0];
    for (int f2 = 0; f2 < 16; ++f2) {
      float h = be1[f2];
      for (int f = 0; f < 16; ++f) h += o[f] * We1[f*16 + f2];
      h = h / (1.f + expf(-h));
      acce += h * We2[f2];
    }
    e = acce;
  }
  out[n] = e;
}

extern "C" void kernel_launch(void* const* d_in, const int* in_sizes, int n_in,
                              void* d_out, int out_size, void* d_ws, size_t ws_size,
                              hipStream_t stream) {
  const float* Rij = (const float*)d_in[0];
  const int* gi = (const int*)d_in[1];
  const int* gj = (const int*)d_in[2];
  const int* Zi = (const int*)d_in[3];
  const unsigned char* pm = (const unsigned char*)d_in[5];
  const unsigned char* nm = (const unsigned char*)d_in[6];
  const float* embed = (const float*)d_in[7];
  const float* W1 = (const float*)d_in[8];
  const float* b1 = (const float*)d_in[9];
  const float* W2 = (const float*)d_in[10];
  const float* b2 = (const float*)d_in[11];
  const float* Wtr = (const float*)d_in[12];
  const float* btr = (const float*)d_in[13];
  const float* ln1s = (const float*)d_in[14];
  const float* ln1b = (const float*)d_in[15];
  const float* Wq1 = (const float*)d_in[16];
  const float* bq1 = (const float*)d_in[17];
  const float* Wq2 = (const float*)d_in[18];
  const float* bq2 = (const float*)d_in[19];
  const float* W3 = (const float*)d_in[20];
  const float* b3 = (const float*)d_in[21];
  const float* ln2s = (const float*)d_in[22];
  const float* ln2b = (const float*)d_in[23];
  const float* We1 = (const float*)d_in[24];
  const float* be1 = (const float*)d_in[25];
  const float* We2 = (const float*)d_in[26];
  const float* be2 = (const float*)d_in[27];
  float* ws = (float*)d_ws;
  float* out = (float*)d_out;

  int tiles = (NPAIR + 15) / 16;          // 6250
  int tb = (tiles + 7) / 8;               // 782 blocks of 8 waves

  k_species  <<<(NNODE + 255) / 256, 256, 0, stream>>>(Zi, nm, embed, ws);
  k_nodestart<<<(NNODE + 1 + 255) / 256, 256, 0, stream>>>(gi, ws);
  k_pair     <<<(NPAIR + 255) / 256, 256, 0, stream>>>(Rij, pm, ws);
  k_nbr      <<<(NNODE + 255) / 256, 256, 0, stream>>>(ws);
  k_ps       <<<(NPAIR + 255) / 256, 256, 0, stream>>>(gi, pm, ws);
  k_stageA   <<<tb, 256, 0, stream>>>(gi, gj, W1, b1, ws);
  k_sumA     <<<(NNODE * 16 + 255) / 256, 256, 0, stream>>>(nm, ws);
  k_stageB   <<<tb, 256, 0, stream>>>(gi, gj, W2, b2, ws);
  k_sphf     <<<(NNODE * 250 + 255) / 256, 256, 0, stream>>>(nm, ws);
  k_node1    <<<(NNODE + 255) / 256, 256, 0, stream>>>(nm, Wtr, btr, ln1s, ln1b, Wq1, bq1, Wq2, bq2, ws, out);
  k_stageC   <<<tb, 256, 0, stream>>>(gi, gj, W3, b3, ws);
  k_node2    <<<(NNODE + 255) / 256, 256, 0, stream>>>(nm, ln2s, ln2b, We1, be1, We2, be2, ws, out);
}